// SoftDiceLoss_43989055045728
// MI455X (gfx1250) — compile-verified
//
#include <hip/hip_runtime.h>

// Soft Dice loss, gfx1250 (MI455X), wave32.
//
// Per batch b:  G = P (16 x HW) x Onehot (HW x 16)  via V_WMMA_F32_16X16X4_F32
//   diag(G)   = overlap[c]
//   rowsum(G) = i_sum[c]           (onehot rows sum to 1)
//   T = Onehot^T x Ones            -> every column = t_sum[c]
//
// Key trick: the 4 k-slots of one WMMA need not be contiguous pixels — lane
// half 0 (k=0,1) and lane half 1 (k=2,3) stream DISJOINT pixel ranges, so each
// lane does contiguous float4/int4 (B128) loads; .xy/.zw feed two WMMAs.
// Memory-bound: 136 MiB @ 23.3 TB/s ~ 6 us floor; WMMA makes the reduction free.

typedef float v2f __attribute__((ext_vector_type(2)));
typedef float v4f __attribute__((ext_vector_type(4)));
typedef float v8f __attribute__((ext_vector_type(8)));
typedef int   v4i __attribute__((ext_vector_type(4)));

#define NB 8
#define NC 16
#define HW (512 * 512)

#define TPB 256
#define WAVES_PER_BLOCK (TPB / 32)
#define BLOCKS_PER_BATCH 32
#define WAVES_PER_BATCH (WAVES_PER_BLOCK * BLOCKS_PER_BATCH)  // 256
#define PIX_PER_WAVE (HW / WAVES_PER_BATCH)                   // 1024
#define PIX_PER_HALF (PIX_PER_WAVE / 2)                       // 512
#define ITERS (PIX_PER_HALF / 8)                              // 64 (8 px/half/iter)

__global__ void dice_zero_kernel(float* __restrict__ ws) {
    // zero G (8*256) + T (8*256) partial-sum matrices = 4096 floats
    ws[blockIdx.x * TPB + threadIdx.x] = 0.0f;
}

__device__ __forceinline__ v8f wmma_f32(v2f a, v2f b, v8f c) {
    return __builtin_amdgcn_wmma_f32_16x16x4_f32(false, a, false, b,
                                                 (short)0, c, false, false);
}

__global__ __launch_bounds__(TPB) void dice_main_kernel(
    const float* __restrict__ pred,   // [NB][NC][HW]
    const int*   __restrict__ tgt,    // [NB][HW]
    float* __restrict__ Gacc,         // [NB][16][16]
    float* __restrict__ Tacc)         // [NB][16][16]
{
    const int b    = blockIdx.x / BLOCKS_PER_BATCH;
    const int blk  = blockIdx.x % BLOCKS_PER_BATCH;
    const int lane = threadIdx.x & 31;
    const int wave = threadIdx.x >> 5;
    const int wgid = blk * WAVES_PER_BLOCK + wave;     // 0..255 within batch

    const int m = lane & 15;     // class row (A) == class col (B)
    const int h = lane >> 4;     // lane half: k=0,1 (h=0) or k=2,3 (h=1)

    // each half streams its own contiguous 512-pixel range
    const int q0 = wgid * PIX_PER_WAVE + h * PIX_PER_HALF;

    const float* __restrict__ ap = pred + (size_t)b * NC * HW + (size_t)m * HW + q0;
    const int*   __restrict__ tp = tgt + (size_t)b * HW + q0;

    v8f g0 = {}; v8f g1 = {};   // overlap/i_sum GEMM partials (2 chains for ILP)
    v8f t0 = {}; v8f t1 = {};   // t_count GEMM partials
    const v2f ones = {1.0f, 1.0f};

    for (int it = 0; it < ITERS; ++it) {
        const int q = it * 8;

        // ---- group 0: this half's pixels q .. q+3 (chain 0) ----
        {
            const v4f f  = __builtin_nontemporal_load((const v4f*)(ap + q));
            const v4i tv = __builtin_nontemporal_load((const v4i*)(tp + q));
            v2f aE = {f.x, f.y};
            v2f aO = {f.z, f.w};
            v2f ohE = {(tv.x == m) ? 1.0f : 0.0f, (tv.y == m) ? 1.0f : 0.0f};
            v2f ohO = {(tv.z == m) ? 1.0f : 0.0f, (tv.w == m) ? 1.0f : 0.0f};
            g0 = wmma_f32(aE, ohE, g0);
            t0 = wmma_f32(ohE, ones, t0);
            g0 = wmma_f32(aO, ohO, g0);
            t0 = wmma_f32(ohO, ones, t0);
        }
        // ---- group 1: this half's pixels q+4 .. q+7 (chain 1) ----
        {
            const v4f f  = __builtin_nontemporal_load((const v4f*)(ap + q + 4));
            const v4i tv = __builtin_nontemporal_load((const v4i*)(tp + q + 4));
            v2f aE = {f.x, f.y};
            v2f aO = {f.z, f.w};
            v2f ohE = {(tv.x == m) ? 1.0f : 0.0f, (tv.y == m) ? 1.0f : 0.0f};
            v2f ohO = {(tv.z == m) ? 1.0f : 0.0f, (tv.w == m) ? 1.0f : 0.0f};
            g1 = wmma_f32(aE, ohE, g1);
            t1 = wmma_f32(ohE, ones, t1);
            g1 = wmma_f32(aO, ohO, g1);
            t1 = wmma_f32(ohO, ones, t1);
        }
    }

    v8f g = g0 + g1;
    v8f t = t0 + t1;

    // C/D layout: lanes 0-15 -> N=lane, VGPR r -> M=r ; lanes 16-31 -> N=lane-16, M=8+r
    float* Gb = Gacc + b * 256;
    float* Tb = Tacc + b * 256;
    const int n     = lane & 15;
    const int mbase = (lane >> 4) << 3;
#pragma unroll
    for (int r = 0; r < 8; ++r) {
        atomicAdd(&Gb[(mbase + r) * 16 + n], g[r]);
        atomicAdd(&Tb[(mbase + r) * 16 + n], t[r]);
    }
}

__global__ void dice_final_kernel(const float* __restrict__ Gacc,
                                  const float* __restrict__ Tacc,
                                  float* __restrict__ out)
{
    const int b = threadIdx.x;
    if (b >= NB) return;
    const float* Gb = Gacc + b * 256;
    const float* Tb = Tacc + b * 256;
    float acc = 0.0f;
    for (int c = 0; c < NC; ++c) {
        const float ov = Gb[c * 16 + c];                    // overlap
        float is = 0.0f;
        for (int n = 0; n < 16; ++n) is += Gb[c * 16 + n];  // i_sum (row sum)
        const float ts = Tb[c * 16];                        // t_sum (any column)
        acc += (2.0f * ov + 1.0f) / (is + ts + 1.0f);
    }
    out[b] = -acc / (float)NC;
}

extern "C" void kernel_launch(void* const* d_in, const int* in_sizes, int n_in,
                              void* d_out, int out_size, void* d_ws, size_t ws_size,
                              hipStream_t stream) {
    (void)in_sizes; (void)n_in; (void)out_size; (void)ws_size;
    const float* pred = (const float*)d_in[0];
    const int*   tgt  = (const int*)d_in[1];
    float* out  = (float*)d_out;
    float* Gacc = (float*)d_ws;          // [8][16][16]
    float* Tacc = Gacc + NB * 256;       // [8][16][16]  (16 KB total scratch)

    dice_zero_kernel<<<(2 * NB * 256) / TPB, TPB, 0, stream>>>(Gacc);
    dice_main_kernel<<<NB * BLOCKS_PER_BATCH, TPB, 0, stream>>>(pred, tgt, Gacc, Tacc);
    dice_final_kernel<<<1, 32, 0, stream>>>(Gacc, Tacc, out);
}